// EnhancedCannyEdgeDetectionModule_816043786652
// MI455X (gfx1250) — compile-verified
//
#include <hip/hip_runtime.h>
#include <hip/hip_bf16.h>
#include <math.h>

// Fused Canny for MI455X (gfx1250, wave32).
// Bandwidth-bound: 134 MB compulsory traffic -> ~5.8us floor @ 23.3 TB/s.
// Single-kernel fusion removes ~400MB of intermediate HBM traffic vs the JAX
// graph. Separable 5x5 Gaussian runs on the matrix pipe as exact-fp32
// V_WMMA_F32_16X16X4_F32 banded-matrix multiplies (5 chained K=4 WMMAs = K=20).
//
// Reference quirk exploited (exact, not approximate): Z = NMS bool in {0,1},
// so weak = (Z>=0.2)&(Z<=0.6) is identically False and out == Z.
//
// Round-1 fix: band weights are now built with a register select chain
// (v_cndmask), not a runtime-indexed array -- the previous version spilled
// w5[] to memory and emitted an exec-masked global_load per WMMA operand
// element. Wave id goes through readfirstlane so the tile loop is scalar.

typedef __attribute__((ext_vector_type(2))) float v2f;
typedef __attribute__((ext_vector_type(8))) float v8f;

#define TS    32   // output tile (32x32)
#define GH    40   // gray tile with +/-4 halo
#define GSTR  48   // gray LDS row stride
#define VSTR  50   // vblur LDS row stride (48 rows; 50 => conflict-free strided A reads)
#define BSTR  40   // blurred LDS row stride (36 rows used)
#define MSTR  36   // mag LDS row stride (34 rows used)

__global__ __launch_bounds__(256)
void canny_fused_kernel(const float* __restrict__ x, float* __restrict__ out,
                        int N, int H, int W)
{
    __shared__ float s_gray[GH * GSTR];     // 7.5 KB  gray, halo 4 (rel rows/cols -4..35)
    __shared__ float s_vb  [48 * VSTR];     // 9.4 KB  vertical-blur (rel -2..45 x -4..43)
    __shared__ float s_bl  [36 * BSTR];     // 5.6 KB  blurred (rel -2..33 x -2..33)
    __shared__ float s_mag [34 * MSTR];     // 4.8 KB  |grad| (rel -1..32 x -1..32)
    __shared__ unsigned char s_cls[TS * TS];// 1 KB    direction class per center pixel

    const int tid  = threadIdx.x;
    const int lane = tid & 31;
    const int wave = __builtin_amdgcn_readfirstlane(tid >> 5); // scalar wave id
    const int half = lane >> 4;   // WMMA lane-half (selects K/M upper half, ISA 7.12.2)
    const int l16  = lane & 15;

    const int tx0 = blockIdx.x * TS;
    const int ty0 = blockIdx.y * TS;
    const int n   = blockIdx.z;

    // ---- Banded Gaussian weight fragments, built once, kept in VGPRs. ----
    // Fragment element (t,v) for this lane = w[K - M] with K = 4t+v+2*half,
    // M = l16, w = normalized 1D Gaussian (sigma=1, size 5). Used as the A
    // operand of the vertical pass AND the B operand of the horizontal pass
    // (both are the same 20x16 / 16x20 band matrix in lane layout).
    v2f bandw[5];
    #pragma unroll
    for (int t = 0; t < 5; ++t) {
        #pragma unroll
        for (int v = 0; v < 2; ++v) {
            int k = 4 * t + v + 2 * half - l16;
            float wv = 0.0f;
            wv = (k == 0) ? 0.05448868f : wv;
            wv = (k == 1) ? 0.24420134f : wv;
            wv = (k == 2) ? 0.40261996f : wv;
            wv = (k == 3) ? 0.24420134f : wv;
            wv = (k == 4) ? 0.05448868f : wv;
            bandw[t][v] = wv;
        }
    }

    // ---------------- Phase 1: gray halo tile (zero outside image == zero-pad conv)
    {
        const float* xr = x + (size_t)n * 3u * H * W;
        const size_t plane = (size_t)H * W;
        for (int i = tid; i < GH * GH; i += 256) {
            int ly = i / GH, lx = i - ly * GH;
            int gy = ty0 + ly - 4, gx = tx0 + lx - 4;
            float g = 0.f;
            if ((unsigned)gy < (unsigned)H && (unsigned)gx < (unsigned)W) {
                size_t p = (size_t)gy * W + gx;
                g = 0.299f * xr[p] + 0.587f * xr[plane + p] + 0.114f * xr[2 * plane + p];
            }
            s_gray[ly * GSTR + lx] = g;
        }
    }
    __syncthreads();

    // ---------------- Phase 2: vertical blur as WMMA: D(16x16) = Band(16x20) x Gray(20x16)
    // 3x3 grid of 16x16 tiles covers rel rows -2..45, cols -4..43
    // (only -2..33 x -4..35 is valid and consumed downstream).
    for (int T = wave; T < 9; T += 8) {           // scalar loop: EXEC stays all-ones
        int tr = T / 3, tc = T - tr * 3;
        int vr0 = -2 + 16 * tr;                   // output-row origin (rel to tile)
        int vc0 = -4 + 16 * tc;                   // column origin
        int colr = vc0 + l16 + 4;                 // gray LDS col (clamp; clamped cells unused)
        colr = colr < 0 ? 0 : (colr > GH - 1 ? GH - 1 : colr);
        v8f acc = {};
        #pragma unroll
        for (int t = 0; t < 5; ++t) {             // 5 chained K=4 WMMAs == K=20
            v2f b;
            #pragma unroll
            for (int v = 0; v < 2; ++v) {
                int j = 4 * t + v + 2 * half;     // absolute K index 0..19
                int row = vr0 + j + 2;            // gray LDS row = (vr0-2+j)+4, clamped
                row = row < 0 ? 0 : (row > GH - 1 ? GH - 1 : row);
                b[v] = s_gray[row * GSTR + colr]; // B[K=j][N=l16]
            }
            acc = __builtin_amdgcn_wmma_f32_16x16x4_f32(
                false, bandw[t], false, b, (short)0, acc, false, false);
        }
        #pragma unroll
        for (int v = 0; v < 8; ++v) {             // D[M=v+8*half][N=l16]
            int row = vr0 + v + 8 * half + 2;     // 0..47
            int col = vc0 + l16 + 4;              // 0..47
            s_vb[row * VSTR + col] = acc[v];
        }
    }
    __syncthreads();

    // ---------------- Phase 3: horizontal blur as WMMA: D = VBlur(16x20) x Band(20x16)
    // 3x3 tiles cover rel -2..45 x -2..45; keep only -2..33 x -2..33.
    for (int T = wave; T < 9; T += 8) {
        int tr = T / 3, tc = T - tr * 3;
        int hr0 = -2 + 16 * tr;
        int hc0 = -2 + 16 * tc;
        int arow = hr0 + l16 + 2;                 // vblur LDS row, 0..47
        v8f acc = {};
        #pragma unroll
        for (int t = 0; t < 5; ++t) {
            v2f a;
            #pragma unroll
            for (int v = 0; v < 2; ++v) {
                int j = 4 * t + v + 2 * half;     // absolute K index 0..19
                int col = hc0 + j + 2;            // vblur LDS col = (hc0-2+j)+4
                col = col > 47 ? 47 : col;        // clamp; clamped cells feed unused outputs
                a[v] = s_vb[arow * VSTR + col];   // A[M=l16][K=j]
            }
            acc = __builtin_amdgcn_wmma_f32_16x16x4_f32(
                false, a, false, bandw[t], (short)0, acc, false, false);
        }
        #pragma unroll
        for (int v = 0; v < 8; ++v) {
            int row = hr0 + v + 8 * half + 2;     // 0..47
            int col = hc0 + l16 + 2;              // 0..47
            if (row < 36 && col < 36)
                s_bl[row * BSTR + col] = acc[v];
        }
    }
    __syncthreads();

    // ---------------- Phase 4: Sobel (cross-correlation, zero-pad) + |grad| + angle class
    for (int i = tid; i < 34 * 34; i += 256) {
        int my = i / 34, mx = i - my * 34;
        int ry = my - 1, rx = mx - 1;             // rel pixel, -1..32
        int gy = ty0 + ry, gx = tx0 + rx;         // image coords
        float b[3][3];
        #pragma unroll
        for (int dy = 0; dy < 3; ++dy)
            #pragma unroll
            for (int dx = 0; dx < 3; ++dx) {
                float v = s_bl[(ry + dy + 1) * BSTR + (rx + dx + 1)];
                // sobel conv zero-pads blurred at image borders
                if ((unsigned)(gy + dy - 1) >= (unsigned)H ||
                    (unsigned)(gx + dx - 1) >= (unsigned)W) v = 0.f;
                b[dy][dx] = v;
            }
        float gxs = (b[0][2] - b[0][0]) + 2.f * (b[1][2] - b[1][0]) + (b[2][2] - b[2][0]);
        float gys = (b[2][0] - b[0][0]) + 2.f * (b[2][1] - b[0][1]) + (b[2][2] - b[0][2]);
        float mag = sqrtf(gxs * gxs + gys * gys);
        bool inImg = (unsigned)gy < (unsigned)H && (unsigned)gx < (unsigned)W;
        s_mag[my * MSTR + mx] = inImg ? mag : 0.f;  // matches jnp.pad(mag, ..., 0)

        if ((unsigned)ry < TS && (unsigned)rx < TS) {
            float ang = atan2f(gys, gxs) * 57.29577951308232f; // 180/pi
            if (ang < 0.f) ang += 180.f;
            int c = (ang < 22.5f) ? 0 : (ang < 67.5f) ? 1 :
                    (ang < 112.5f) ? 2 : (ang < 157.5f) ? 3 : 0; // [157.5,180] -> class 0
            s_cls[ry * TS + rx] = (unsigned char)c;
        }
    }
    __syncthreads();

    // ---------------- Phase 5: NMS -> out (== Z; weak-mask is provably always false)
    {
        float* outn = out + (size_t)n * H * W;
        for (int i = tid; i < TS * TS; i += 256) {
            int ry = i / TS, rx = i - ry * TS;
            int my = ry + 1, mx = rx + 1;
            const float* mrow = &s_mag[my * MSTR + mx];
            float m  = mrow[0];
            float lf = mrow[-1],        rt = mrow[1];
            float up = mrow[-MSTR],     dn = mrow[MSTR];
            float ul = mrow[-MSTR - 1], ur = mrow[-MSTR + 1];
            float dl = mrow[ MSTR - 1], dr = mrow[ MSTR + 1];
            int c = s_cls[i];
            bool Z = (c == 0) ? (m >= lf && m >= rt)
                   : (c == 1) ? (m >= dl && m >= ur)
                   : (c == 2) ? (m >= up && m >= dn)
                   :            (m >= ul && m >= dr);
            outn[(size_t)(ty0 + ry) * W + (tx0 + rx)] = Z ? 1.0f : 0.0f;
        }
    }
}

extern "C" void kernel_launch(void* const* d_in, const int* in_sizes, int n_in,
                              void* d_out, int out_size, void* d_ws, size_t ws_size,
                              hipStream_t stream) {
    (void)n_in; (void)out_size; (void)d_ws; (void)ws_size;
    const float* x = (const float*)d_in[0];
    float* out = (float*)d_out;
    const int H = 512, W = 512;
    const int N = in_sizes[0] / (3 * H * W);      // 32
    dim3 grid(W / TS, H / TS, N);                 // 16 x 16 x 32 = 8192 blocks
    canny_fused_kernel<<<grid, 256, 0, stream>>>(x, out, N, H, W);
}